// l_feature_extract_layer_7395933684255
// MI455X (gfx1250) — compile-verified
//
#include <hip/hip_runtime.h>

// f32 WMMA operand types: 16x16x4 f32 -> A,B are 2 VGPRs/lane, C/D are 8 VGPRs/lane
typedef float v2f __attribute__((ext_vector_type(2)));
typedef float v8f __attribute__((ext_vector_type(8)));

#define C_CLASSES 10
#define D_DIM     256
#define V_VIEWS   2
#define NTILES    32          // V(2) * (D/16 = 16) output tiles of 16x16
#define TILE_ELEMS 256        // 16x16 floats per tile
#define WAVES_PER_BLOCK 8

// Stage 1: each wave computes a partial 16x16 tile (classes x d-columns) for
// its N-split using chained V_WMMA_F32_16X16X4_F32 (D = A*B + C accumulation).
// K is unrolled 4x (16 nodes/iter) so all loads cluster into one clause and
// the wave issues 4 WMMAs back-to-back per memory wait.
__global__ __launch_bounds__(256)
void masked_sum_stage1(const int* __restrict__ y,
                       const float* __restrict__ x,
                       float* __restrict__ partial,
                       int N, int chunk)
{
    const int lane  = threadIdx.x & 31;
    const int wave  = threadIdx.x >> 5;
    const int tile  = blockIdx.x * WAVES_PER_BLOCK + wave;   // 0..31
    const int v     = tile >> 4;                              // view
    const int dtile = tile & 15;                              // 16-wide d stripe
    const int d0    = dtile * 16;
    const int split = blockIdx.y;

    long nStart = (long)split * chunk;
    long nEnd   = nStart + chunk;
    if (nEnd > N) nEnd = N;

    const int c      = lane & 15;          // class row (M) for A
    const int col    = lane & 15;          // d column (N) for B
    const int koff   = (lane >> 4) * 2;    // lanes 0-15 -> K0/K1, lanes 16-31 -> K2/K3
    const int cclamp = (c < C_CLASSES) ? c : (C_CLASSES - 1);   // in-bounds always
    const float cm   = (c < C_CLASSES) ? 1.0f : 0.0f;           // zero pad rows c>=10

    // Per-lane base pointers: unconditional, branch-free loads from here on.
    const int*   __restrict__ yrow = y + cclamp;                        // + n*10
    const float* __restrict__ xcol = x + (size_t)v * (size_t)N * D_DIM + d0 + col; // + n*256

    v8f acc = {0.f, 0.f, 0.f, 0.f, 0.f, 0.f, 0.f, 0.f};

    long n = nStart;
    // Main loop: 16 nodes (4 WMMA-K-chunks) per iteration.
    for (; n + 16 <= nEnd; n += 16) {
        v2f a[4], b[4];
        #pragma unroll
        for (int u = 0; u < 4; ++u) {
            const long na0 = n + 4 * u + koff;        // K row pair base
            const int yv0 = yrow[na0 * C_CLASSES];
            const int yv1 = yrow[na0 * C_CLASSES + C_CLASSES];
            b[u].x = xcol[na0 * D_DIM];
            b[u].y = xcol[na0 * D_DIM + D_DIM];
            a[u].x = (yv0 == 0) ? cm : 0.0f;
            a[u].y = (yv1 == 0) ? cm : 0.0f;
        }
        #pragma unroll
        for (int u = 0; u < 4; ++u)
            acc = __builtin_amdgcn_wmma_f32_16x16x4_f32(
                      false, a[u], false, b[u], (short)0, acc, false, false);
    }
    // Tail: single K=4 chunks (branch-free loads as above).
    for (; n < nEnd; n += 4) {
        const long na0 = n + koff;
        const int yv0 = yrow[na0 * C_CLASSES];
        const int yv1 = yrow[na0 * C_CLASSES + C_CLASSES];
        v2f a, b;
        b.x = xcol[na0 * D_DIM];
        b.y = xcol[na0 * D_DIM + D_DIM];
        a.x = (yv0 == 0) ? cm : 0.0f;
        a.y = (yv1 == 0) ? cm : 0.0f;
        acc = __builtin_amdgcn_wmma_f32_16x16x4_f32(
                  false, a, false, b, (short)0, acc, false, false);
    }

    // Store tile row-major [m][nn]: VGPR r of lane L holds element
    // (M = r + 8*(L>=16), N = L&15) per the CDNA5 16x16 f32 C/D layout.
    float* __restrict__ pt = partial + ((size_t)split * NTILES + tile) * TILE_ELEMS;
    const int mbase = (lane >> 4) * 8;
    #pragma unroll
    for (int r = 0; r < 8; ++r)
        pt[(mbase + r) * 16 + col] = acc[r];
}

// Stage 2: deterministic fixed-order reduction over splits; writes all of d_out.
__global__ void masked_sum_stage2(const float* __restrict__ partial,
                                  float* __restrict__ out, int nsplit)
{
    const int idx = blockIdx.x * blockDim.x + threadIdx.x;   // 0..V*C*D-1
    if (idx >= V_VIEWS * C_CLASSES * D_DIM) return;
    const int v   = idx / (C_CLASSES * D_DIM);
    const int cd  = idx % (C_CLASSES * D_DIM);
    const int cls = cd / D_DIM;
    const int d   = cd % D_DIM;
    const int tile = v * 16 + (d >> 4);
    const int elem = cls * 16 + (d & 15);

    float s = 0.f;
    for (int sp = 0; sp < nsplit; ++sp)
        s += partial[((size_t)sp * NTILES + tile) * TILE_ELEMS + elem];
    out[idx] = s;
}

extern "C" void kernel_launch(void* const* d_in, const int* in_sizes, int n_in,
                              void* d_out, int out_size, void* d_ws, size_t ws_size,
                              hipStream_t stream) {
    const int*   y = (const int*)d_in[0];      // [N, 10] int32
    const float* x = (const float*)d_in[1];    // [2, N, 256] f32
    float* out     = (float*)d_out;            // [2, 2560] f32
    float* partial = (float*)d_ws;

    const int N = in_sizes[0] / C_CLASSES;

    // Choose split count from available scratch (each split needs 32 tiles * 1KB).
    const size_t bytesPerSplit = (size_t)NTILES * TILE_ELEMS * sizeof(float);
    long nsplit = (long)(ws_size / bytesPerSplit);
    if (nsplit < 1)   nsplit = 1;
    if (nsplit > 256) nsplit = 256;

    // Chunk must be a multiple of 16 (4x-unrolled WMMA K); recompute split count.
    long chunk = (N + nsplit - 1) / nsplit;
    chunk = (chunk + 15) & ~15L;
    const int splits = (int)((N + chunk - 1) / chunk);

    dim3 grid1(NTILES / WAVES_PER_BLOCK, splits);
    masked_sum_stage1<<<grid1, 256, 0, stream>>>(y, x, partial, N, (int)chunk);

    const int nout = V_VIEWS * C_CLASSES * D_DIM;
    masked_sum_stage2<<<(nout + 255) / 256, 256, 0, stream>>>(partial, out, splits);
}